// YOLOV7_86629490360415
// MI455X (gfx1250) — compile-verified
//
#include <hip/hip_runtime.h>
#include <cstdint>

// YOLOv7 P3 head decode, MI455X (gfx1250).
// Memory-bound: 208.8 MB HBM traffic, ~9.0 us floor at 23.3 TB/s.
// Strategy: LDS transpose staging via async global->LDS (ASYNCcnt path),
// conflict-free banking, b128 coalesced NT stores.

#define NB      16
#define NA      3
#define NCH     85           // 5 + 80 classes
#define WW      80
#define HWTOT   6400         // 80*80
#define TP      128          // positions per tile
#define TILES   (HWTOT / TP) // 50
#define TILE_ELEMS (NCH * TP) // 10880 floats = 43,520 B LDS
#define THREADS 256

typedef float floatx4 __attribute__((ext_vector_type(4)));

__device__ __forceinline__ float fast_sigmoid(float v) {
    // 1 / (1 + exp(-v)) via native exp2 + rcp
    float e = __builtin_amdgcn_exp2f(v * -1.44269504088896341f);
    return __builtin_amdgcn_rcpf(1.0f + e);
}

__global__ __launch_bounds__(THREADS)
void yolo_p3_head(const float* __restrict__ in,
                  const float* __restrict__ anchors,
                  float* __restrict__ out) {
    // Transposed tile: lds flat index = pos*85 + ch  (== output flat index)
    __shared__ __align__(16) float tile[TILE_ELEMS];

    const int tid = threadIdx.x;
    const int blk = blockIdx.x;
    const int t   = blk % TILES;
    const int a   = (blk / TILES) % NA;
    const int b   = blk / (TILES * NA);
    const int p0  = t * TP;

    const float aw = anchors[2 * a + 0];
    const float ah = anchors[2 * a + 1];

    const float* inbase = in + ((size_t)b * (NA * NCH) + (size_t)a * NCH) * HWTOT + p0;

    // ---- Stage 1: global -> LDS, transposing channel-major -> position-major.
    // Lane mapping: i -> (ch = i>>7, pos = i&127). Global: each wave reads 128
    // contiguous bytes (fully coalesced). LDS: lane stride = 85 dwords (odd) ->
    // gcd(85,64)=1 -> conflict-free bank pattern. Async path: no VGPR
    // round-trip, tracked on ASYNCcnt.
    for (int i = tid; i < TILE_ELEMS; i += THREADS) {
        const int ch  = i >> 7;
        const int pos = i & (TP - 1);
        const unsigned long long gaddr =
            (unsigned long long)(uintptr_t)(inbase + (size_t)ch * HWTOT + pos);
        const unsigned lofs = (unsigned)(uintptr_t)&tile[pos * NCH + ch];
        asm volatile("global_load_async_to_lds_b32 %0, %1, off"
                     :
                     : "v"(lofs), "v"(gaddr)
                     : "memory");
    }
#if __has_builtin(__builtin_amdgcn_s_wait_asynccnt)
    __builtin_amdgcn_s_wait_asynccnt(0);
#else
    asm volatile("s_wait_asynccnt 0" ::: "memory");
#endif
    __syncthreads();

    // ---- Stage 2: decode + fully-coalesced b128 NT stores.
    // Output tile is one contiguous 43,520 B region; outbase is 16B-aligned.
    float* outbase = out + ((size_t)b * (NA * HWTOT) + (size_t)a * HWTOT + p0) * NCH;

    for (int k = tid; k < TILE_ELEMS / 4; k += THREADS) {
        const int j0 = k * 4;
        const floatx4 v4 = *(const floatx4*)&tile[j0];  // conflict-free ds_load_b128
        floatx4 o4;
#pragma unroll
        for (int r = 0; r < 4; ++r) {
            const unsigned j   = (unsigned)(j0 + r);
            const unsigned pos = j / NCH;            // const-div -> mul/shift
            const unsigned ch  = j - pos * NCH;
            const float v = v4[r];
            const float s = fast_sigmoid(v);
            float o = s;                              // conf + classes path
            if (ch < 4u) {
                const unsigned p = (unsigned)p0 + pos;
                if (ch == 0u) {
                    o = (s + (float)(p % WW)) * 8.0f;        // px
                } else if (ch == 1u) {
                    o = (s + (float)(p / WW)) * 8.0f;        // py
                } else {
                    const float c = fminf(fmaxf(v, -16.0f), 16.0f);
                    const float e = __builtin_amdgcn_exp2f(c * 1.44269504088896341f);
                    o = e * ((ch == 2u) ? aw : ah);          // pw / ph
                }
            }
            o4[r] = o;
        }
        // Write-once output: non-temporal to avoid polluting L2 with 104 MB.
        __builtin_nontemporal_store(o4, (floatx4*)(outbase + j0));
    }
}

extern "C" void kernel_launch(void* const* d_in, const int* in_sizes, int n_in,
                              void* d_out, int out_size, void* d_ws, size_t ws_size,
                              hipStream_t stream) {
    (void)in_sizes; (void)n_in; (void)out_size; (void)d_ws; (void)ws_size;
    const float* in      = (const float*)d_in[0];
    const float* anchors = (const float*)d_in[1];
    float* out           = (float*)d_out;

    const dim3 grid(NB * NA * TILES);   // 2400 blocks
    const dim3 block(THREADS);          // 8 wave32 waves
    hipLaunchKernelGGL(yolo_p3_head, grid, block, 0, stream, in, anchors, out);
}